// LinearCrossAttention_77197742178857
// MI455X (gfx1250) — compile-verified
//
#include <hip/hip_runtime.h>

typedef __attribute__((ext_vector_type(2))) float v2f;
typedef __attribute__((ext_vector_type(8))) float v8f;

#define C 256
#define HW 128
#define NSP 16384            // HW*HW
#define BATCH 4
#define BCN (BATCH * C * NSP)
#define BN (BATCH * NSP)

__device__ __forceinline__ float delu(float x) {
    // param*relu(x) + expm1(param*min(x,0)) + 1  ==  x>0 ? 10x+1 : exp(10x)
    return x > 0.0f ? fmaf(10.0f, x, 1.0f) : __expf(10.0f * x);
}

__device__ __forceinline__ v8f wmma_f32(v2f a, v2f b, v8f c) {
    return __builtin_amdgcn_wmma_f32_16x16x4_f32(false, a, false, b, (short)0, c,
                                                 false, false);
}

// ---------------------------------------------------------------------------
// Kernel 0: fuse 3x3 + 1x1 output weights into wtap[t][o][c], bfused[o]
// ---------------------------------------------------------------------------
__global__ void prep_kernel(const float* __restrict__ wd1, const float* __restrict__ bd1,
                            const float* __restrict__ wd3, const float* __restrict__ bd3,
                            float* __restrict__ wtap, float* __restrict__ bfused) {
    int gid = blockIdx.x * 256 + threadIdx.x;
    if (gid < 9 * C * C) {
        int t = gid / (C * C);          // tap = ky*3+kx
        int oc = gid - t * (C * C);     // o*C + c
        float v = wd3[oc * 9 + t];
        if (t == 4) v += wd1[oc];       // fold 1x1 into center tap
        wtap[gid] = v;
    }
    if (gid < C) bfused[gid] = bd3[gid] + bd1[gid];
}

// ---------------------------------------------------------------------------
// Kernel 1: per-stream projections via f32 WMMA.
//   k   = delu(Wk X + bk)        -> k_out (B,C,N)
//   KV  = k * (Wv X + bv)        -> kv_out (B,C,N)
//   qsum[n] = sum_o delu(Wq X + bq)[o,n]   -> qsum_out (B,N)   (q not stored)
// LDS X tile is channel-pair interleaved: element (c, col) lives at
// Xs[(c>>1)*XS2 + col*2 + (c&1)], so a B operand (K, K+1) is one ds_load_b64.
// ---------------------------------------------------------------------------
#define XS2 66   // dword stride per channel-pair row (32 cols * 2 + pad)
__device__ __forceinline__ void gemm_pass(const float* __restrict__ W,
                                          const float* Xs, int obase, int lh,
                                          int half, v8f acc[2]) {
    acc[0] = v8f{};
    acc[1] = v8f{};
    const float* wrow = W + (obase + lh) * C + 2 * half;
    const float* xb = Xs + half * XS2 + lh * 2;
    for (int c0 = 0; c0 < C; c0 += 4) {
        v2f a = *(const v2f*)(wrow + c0);
        const float* xp = xb + (c0 >> 1) * XS2;
        acc[0] = wmma_f32(a, *(const v2f*)(xp), acc[0]);
        acc[1] = wmma_f32(a, *(const v2f*)(xp + 32), acc[1]);
    }
}

__global__ __launch_bounds__(256) void proj_kernel(
    const float* __restrict__ x,
    const float* __restrict__ wq, const float* __restrict__ bq,
    const float* __restrict__ wk, const float* __restrict__ bk,
    const float* __restrict__ wv, const float* __restrict__ bv,
    float* __restrict__ k_out, float* __restrict__ kv_out,
    float* __restrict__ qsum_out) {
    __shared__ float Xs[(C / 2) * XS2];
    __shared__ float qs[32];
    const int tid = threadIdx.x;
    const int b = blockIdx.y;
    const int n0 = blockIdx.x * 32;
    const int w = tid >> 5;
    const int lane = tid & 31;
    const int half = lane >> 4;
    const int lh = lane & 15;

    if (tid < 32) qs[tid] = 0.0f;
    for (int i = tid; i < (C / 2) * 32; i += 256) {
        int c2 = i >> 5, j = i & 31;
        v2f p;
        p.x = x[(b * C + 2 * c2) * NSP + n0 + j];
        p.y = x[(b * C + 2 * c2 + 1) * NSP + n0 + j];
        *(v2f*)&Xs[c2 * XS2 + j * 2] = p;
    }
    __syncthreads();

    for (int rep = 0; rep < 2; ++rep) {
        const int obase = (w + rep * 8) * 16;
        v8f kacc[2], vacc[2], qacc[2];

        gemm_pass(wk, Xs, obase, lh, half, kacc);
#pragma unroll
        for (int nt = 0; nt < 2; ++nt)
#pragma unroll
            for (int r = 0; r < 8; ++r) {
                int o = obase + r + 8 * half;
                float kv = delu(kacc[nt][r] + bk[o]);
                kacc[nt][r] = kv;
                k_out[(b * C + o) * NSP + n0 + nt * 16 + lh] = kv;
            }

        gemm_pass(wv, Xs, obase, lh, half, vacc);
#pragma unroll
        for (int nt = 0; nt < 2; ++nt)
#pragma unroll
            for (int r = 0; r < 8; ++r) {
                int o = obase + r + 8 * half;
                float vv = vacc[nt][r] + bv[o];
                kv_out[(b * C + o) * NSP + n0 + nt * 16 + lh] = kacc[nt][r] * vv;
            }

        gemm_pass(wq, Xs, obase, lh, half, qacc);
#pragma unroll
        for (int nt = 0; nt < 2; ++nt) {
            float s = 0.0f;
#pragma unroll
            for (int r = 0; r < 8; ++r)
                s += delu(qacc[nt][r] + bq[obase + r + 8 * half]);
            atomicAdd(&qs[nt * 16 + lh], s);   // lanes l and l+16 hit same column
        }
    }
    __syncthreads();
    if (tid < 32) qsum_out[b * NSP + n0 + tid] = qs[tid];
}

// ---------------------------------------------------------------------------
// Kernel 2: esum[b,c] = sum_n qsum[b,n] * k_other[b,c,n]   (block reduction)
// ---------------------------------------------------------------------------
__global__ __launch_bounds__(256) void esum_kernel(
    const float* __restrict__ qs1, const float* __restrict__ k2,
    const float* __restrict__ qs2, const float* __restrict__ k1,
    float* __restrict__ es1, float* __restrict__ es2) {
    const int c = blockIdx.x, b = blockIdx.y, which = blockIdx.z;
    const float* q = which ? qs2 : qs1;
    const float* k = which ? k1 : k2;
    float* e = which ? es2 : es1;
    const float* kr = k + (b * C + c) * NSP;
    const float* qr = q + b * NSP;
    float acc = 0.0f;
    for (int n = threadIdx.x; n < NSP; n += 256) acc += qr[n] * kr[n];
    __shared__ float red[256];
    red[threadIdx.x] = acc;
    __syncthreads();
    for (int s = 128; s > 0; s >>= 1) {
        if (threadIdx.x < s) red[threadIdx.x] += red[threadIdx.x + s];
        __syncthreads();
    }
    if (threadIdx.x == 0) e[b * C + c] = red[0];
}

// ---------------------------------------------------------------------------
// Kernel 3: norm[b,n] = 1 / (sum_c esum[b,c]*k_other[b,c,n] + eps)
// ---------------------------------------------------------------------------
__global__ __launch_bounds__(256) void norm_kernel(
    const float* __restrict__ es1, const float* __restrict__ k2,
    const float* __restrict__ es2, const float* __restrict__ k1,
    float* __restrict__ n1, float* __restrict__ n2) {
    const int b = blockIdx.y, which = blockIdx.z;
    const float* e = which ? es2 : es1;
    const float* k = which ? k1 : k2;
    float* nrm = which ? n2 : n1;
    __shared__ float es[C];
    es[threadIdx.x] = e[b * C + threadIdx.x];
    __syncthreads();
    const int n = blockIdx.x * 256 + threadIdx.x;
    const float* kb = k + (size_t)b * C * NSP + n;
    float acc = 0.0f;
#pragma unroll 4
    for (int c = 0; c < C; ++c) acc += es[c] * kb[c * NSP];
    nrm[b * NSP + n] = 1.0f / (acc + 1e-10f);
}

// ---------------------------------------------------------------------------
// Kernel 4: implicit-GEMM fused 3x3 conv via f32 WMMA + attention epilogue:
//   out = x + gamma * (conv(x) + bfused + esum*KV*norm)
// Workgroup: one (batch, row h, 64-channel o-chunk). 8 waves = 8 n-tiles of 16.
// Halo tile in LDS is channel-pair interleaved (one ds_load_b64 per B operand):
// element (cc, r, col) at xs[(((cc>>1)*3 + r)*132 + col)*2 + (cc&1)].
// ---------------------------------------------------------------------------
__global__ __launch_bounds__(256) void conv_att_kernel(
    const float* __restrict__ x, const float* __restrict__ wtap,
    const float* __restrict__ bfused, const float* __restrict__ esum,
    const float* __restrict__ kv, const float* __restrict__ nrm,
    const float* __restrict__ gamma, float* __restrict__ out) {
    __shared__ float xs[16 * 3 * 132 * 2];   // 50688 B
    const int tid = threadIdx.x;
    const int ochunk = blockIdx.x;   // 0..3 -> 64 output channels
    const int h = blockIdx.y;        // 0..127
    const int b = blockIdx.z;        // 0..3
    const int w = tid >> 5;          // wave id = n-tile
    const int lane = tid & 31;
    const int half = lane >> 4;
    const int lh = lane & 15;
    const int wl = w * 16;

    v8f acc[4] = {v8f{}, v8f{}, v8f{}, v8f{}};

    for (int cchunk = 0; cchunk < 8; ++cchunk) {
        const int cbase = cchunk * 32;
        __syncthreads();
        for (int i = tid; i < 16 * 3 * 130; i += 256) {
            int c2 = i / 390;
            int rem = i - c2 * 390;
            int r = rem / 130;
            int col = rem - r * 130;
            int gh = h + r - 1;
            int gw = col - 1;
            v2f p;
            p.x = 0.0f;
            p.y = 0.0f;
            if (gh >= 0 && gh < HW && gw >= 0 && gw < HW) {
                int base = (b * C + cbase + 2 * c2) * NSP + gh * HW + gw;
                p.x = x[base];
                p.y = x[base + NSP];
            }
            *(v2f*)&xs[((c2 * 3 + r) * 132 + col) * 2] = p;
        }
        __syncthreads();
#pragma unroll
        for (int t = 0; t < 9; ++t) {
            const int ky = t / 3, kx = t - ky * 3;
            const float* wp = wtap + t * C * C + (ochunk * 64 + lh) * C + cbase + 2 * half;
            const float* xrow = xs + ((half * 3 + ky) * 132 + kx + wl + lh) * 2;
            for (int c0 = 0; c0 < 32; c0 += 4) {
                v2f bb = *(const v2f*)(xrow + (c0 >> 1) * 792);  // 792 = 3*132*2
#pragma unroll
                for (int ot = 0; ot < 4; ++ot) {
                    v2f a = *(const v2f*)(wp + ot * 16 * C + c0);
                    acc[ot] = wmma_f32(a, bb, acc[ot]);
                }
            }
        }
    }

    const float g = gamma[0];
    const int n = h * HW + wl + lh;
    const float nv = nrm[b * NSP + n];
#pragma unroll
    for (int ot = 0; ot < 4; ++ot) {
#pragma unroll
        for (int r = 0; r < 8; ++r) {
            const int o = ochunk * 64 + ot * 16 + r + 8 * half;
            const int idx = (b * C + o) * NSP + n;
            const float e = esum[b * C + o];
            out[idx] = x[idx] + g * (acc[ot][r] + bfused[o] + e * kv[idx] * nv);
        }
    }
}

// ---------------------------------------------------------------------------
extern "C" void kernel_launch(void* const* d_in, const int* in_sizes, int n_in,
                              void* d_out, int out_size, void* d_ws, size_t ws_size,
                              hipStream_t stream) {
    (void)in_sizes; (void)n_in; (void)out_size; (void)ws_size;
    const float* x1 = (const float*)d_in[0];
    const float* x2 = (const float*)d_in[1];
    const float* wq1 = (const float*)d_in[2];  const float* bq1 = (const float*)d_in[3];
    const float* wk1 = (const float*)d_in[4];  const float* bk1 = (const float*)d_in[5];
    const float* wv1 = (const float*)d_in[6];  const float* bv1 = (const float*)d_in[7];
    const float* wq2 = (const float*)d_in[8];  const float* bq2 = (const float*)d_in[9];
    const float* wk2 = (const float*)d_in[10]; const float* bk2 = (const float*)d_in[11];
    const float* wv2 = (const float*)d_in[12]; const float* bv2 = (const float*)d_in[13];
    const float* wd1 = (const float*)d_in[14]; const float* bd1 = (const float*)d_in[15];
    const float* wd3 = (const float*)d_in[16]; const float* bd3 = (const float*)d_in[17];
    const float* gm1 = (const float*)d_in[18]; const float* gm2 = (const float*)d_in[19];

    float* ws = (float*)d_ws;
    float* k1   = ws;            float* k2   = k1 + BCN;
    float* kv1  = k2 + BCN;      float* kv2  = kv1 + BCN;
    float* qs1  = kv2 + BCN;     float* qs2  = qs1 + BN;
    float* nrm1 = qs2 + BN;      float* nrm2 = nrm1 + BN;
    float* es1  = nrm2 + BN;     float* es2  = es1 + BATCH * C;
    float* wtap = es2 + BATCH * C;
    float* bfu  = wtap + 9 * C * C;
    float* out1 = (float*)d_out; float* out2 = out1 + BCN;

    prep_kernel<<<(9 * C * C + 255) / 256, 256, 0, stream>>>(wd1, bd1, wd3, bd3, wtap, bfu);

    dim3 gp(NSP / 32, BATCH);
    proj_kernel<<<gp, 256, 0, stream>>>(x1, wq1, bq1, wk1, bk1, wv1, bv1, k1, kv1, qs1);
    proj_kernel<<<gp, 256, 0, stream>>>(x2, wq2, bq2, wk2, bk2, wv2, bv2, k2, kv2, qs2);

    esum_kernel<<<dim3(C, BATCH, 2), 256, 0, stream>>>(qs1, k2, qs2, k1, es1, es2);
    norm_kernel<<<dim3(NSP / 256, BATCH, 2), 256, 0, stream>>>(es1, k2, es2, k1, nrm1, nrm2);

    dim3 gc(4, HW, BATCH);
    conv_att_kernel<<<gc, 256, 0, stream>>>(x1, wtap, bfu, es1, kv1, nrm1, gm1, out1);
    conv_att_kernel<<<gc, 256, 0, stream>>>(x2, wtap, bfu, es2, kv2, nrm2, gm2, out2);
}